// distance2edge_batched_60507499266663
// MI455X (gfx1250) — compile-verified
//
#include <hip/hip_runtime.h>

// Problem constants (match reference)
#define E_TOTAL 33554432          // number of candidate pairs
#define R0f     0.1f              // radius threshold

#define BLOCK   256               // 8 wave32 waves per block
#define IPT     4                 // items per thread (one 128-bit load)
#define CHUNK   (BLOCK * IPT)     // 1024 elements per block
#define NBLK    (E_TOTAL / CHUNK) // 32768 blocks  (E is a multiple of CHUNK)

#define WAVE    32                // gfx1250 is wave32-only (probe-confirmed)
#define NWAVES  (BLOCK / WAVE)    // 8 waves per block

#define SCAN_T  1024
#define PER_T   (NBLK / SCAN_T)   // 32 block-counts per scan thread

// Native Clang vector types: required by __builtin_nontemporal_{load,store}
typedef float v4f __attribute__((ext_vector_type(4)));
typedef int   v4i __attribute__((ext_vector_type(4)));

// Wave32 ballot: low 32 bits hold the mask on gfx1250
__device__ __forceinline__ unsigned ballot32(bool p) {
    return (unsigned)__ballot(p);
}

// ---------------------------------------------------------------------------
// Pass 1: per-block predicate count + fill output with -1 (non-temporal).
// Counting is ballot+popcount per wave (no LDS scan); one barrier total.
// NT stores on the fill keep the 268MB write-once stream out of L2 so the
// 134MB distance stream stays resident (192MB global L2) for pass 3.
// ---------------------------------------------------------------------------
__global__ void k_count_fill(const float* __restrict__ dist,
                             float* __restrict__ out,          // [2E+1]
                             unsigned* __restrict__ blkCnt) {  // [NBLK]
    __shared__ unsigned sWave[NWAVES];
    const int    t    = threadIdx.x;
    const int    lane = t & (WAVE - 1);
    const int    wid  = t >> 5;
    const size_t base = (size_t)blockIdx.x * CHUNK;

    const v4f d4 = reinterpret_cast<const v4f*>(dist + base)[t];

    const unsigned m0 = ballot32(d4.x < R0f);
    const unsigned m1 = ballot32(d4.y < R0f);
    const unsigned m2 = ballot32(d4.z < R0f);
    const unsigned m3 = ballot32(d4.w < R0f);
    const unsigned waveCnt = __popc(m0) + __popc(m1) + __popc(m2) + __popc(m3);

    if (lane == 0) sWave[wid] = waveCnt;
    __syncthreads();
    if (t == 0) {
        unsigned total = 0;
#pragma unroll
        for (int w = 0; w < NWAVES; ++w) total += sWave[w];
        blkCnt[blockIdx.x] = total;
    }

    // Fill this block's src/dst output chunks with -1 (non-temporal b128 stores)
    const v4f m1v = {-1.f, -1.f, -1.f, -1.f};
    v4f* srcChunk = reinterpret_cast<v4f*>(out + base);
    v4f* dstChunk = reinterpret_cast<v4f*>(out + (size_t)E_TOTAL + base);
    __builtin_nontemporal_store(m1v, srcChunk + t);
    __builtin_nontemporal_store(m1v, dstChunk + t);
}

// ---------------------------------------------------------------------------
// Pass 2: exclusive scan of the 32768 block counts (single 1024-thread block).
// Each thread serially scans 32 counts; Hillis-Steele LDS scan of the 1024
// thread-totals stitches them together. Also writes n_edges to out[2E].
// ---------------------------------------------------------------------------
__global__ void k_scan(unsigned* __restrict__ blkCnt,
                       float* __restrict__ out) {
    __shared__ unsigned s[SCAN_T];
    const int t = threadIdx.x;

    unsigned v[PER_T];
    unsigned sum = 0;
#pragma unroll
    for (int i = 0; i < PER_T; ++i) {
        v[i] = blkCnt[t * PER_T + i];
        sum += v[i];
    }
    s[t] = sum;
    __syncthreads();

    for (int off = 1; off < SCAN_T; off <<= 1) {
        unsigned x = (t >= off) ? s[t - off] : 0u;
        __syncthreads();
        s[t] += x;
        __syncthreads();
    }

    unsigned excl = (t == 0) ? 0u : s[t - 1];
#pragma unroll
    for (int i = 0; i < PER_T; ++i) {
        unsigned tmp = v[i];
        blkCnt[t * PER_T + i] = excl;  // overwrite with exclusive prefix
        excl += tmp;
    }
    if (t == SCAN_T - 1) {
        out[2 * (size_t)E_TOTAL] = (float)s[SCAN_T - 1];  // n_edges
    }
}

// ---------------------------------------------------------------------------
// Pass 3: order-preserving scatter. Element 4t+j precedes 4t'+j' iff
// (t,j) < (t',j') lexicographically, so the global rank is:
//   blkOff[b] + (sum of totals of earlier waves)           [1 LDS barrier]
//             + (items of earlier lanes in my wave, any j)  [ballot+popc]
//             + (my own items with smaller j)               [sequential]
// Edge pairs are read once with non-temporal loads (don't evict distances);
// compacted outputs written with non-temporal stores (write-once).
// ---------------------------------------------------------------------------
__global__ void k_scatter(const float* __restrict__ dist,
                          const int* __restrict__ edges,          // [E][2]
                          const unsigned* __restrict__ blkOff,    // [NBLK] exclusive
                          float* __restrict__ out) {
    __shared__ unsigned sWave[NWAVES];
    const int    t    = threadIdx.x;
    const int    lane = t & (WAVE - 1);
    const int    wid  = t >> 5;
    const size_t base = (size_t)blockIdx.x * CHUNK;

    const v4f d4 = reinterpret_cast<const v4f*>(dist + base)[t];
    const bool p0 = d4.x < R0f, p1 = d4.y < R0f, p2 = d4.z < R0f, p3 = d4.w < R0f;

    const unsigned m0 = ballot32(p0);
    const unsigned m1 = ballot32(p1);
    const unsigned m2 = ballot32(p2);
    const unsigned m3 = ballot32(p3);
    const unsigned waveCnt = __popc(m0) + __popc(m1) + __popc(m2) + __popc(m3);

    if (lane == 0) sWave[wid] = waveCnt;
    __syncthreads();

    // Sum of earlier waves' totals (wid is wave-uniform; <=7 LDS loads)
    unsigned waveBase = 0;
    for (int w = 0; w < wid; ++w) waveBase += sWave[w];

    // Items belonging to earlier lanes of this wave (any slot j)
    const unsigned lt = (1u << lane) - 1u;
    const unsigned laneRank = __popc(m0 & lt) + __popc(m1 & lt) +
                              __popc(m2 & lt) + __popc(m3 & lt);

    unsigned pos = blkOff[blockIdx.x] + waveBase + laneRank;

    if (p0 | p1 | p2 | p3) {
        // 4 consecutive (src,dst) int pairs = two 128-bit non-temporal loads
        const v4i* e4 = reinterpret_cast<const v4i*>(edges + 2 * base);
        const v4i eA = __builtin_nontemporal_load(e4 + 2 * t);
        const v4i eB = __builtin_nontemporal_load(e4 + 2 * t + 1);

        float* srcOut = out;
        float* dstOut = out + (size_t)E_TOTAL;
        if (p0) { __builtin_nontemporal_store((float)eA.x, srcOut + pos);
                  __builtin_nontemporal_store((float)eA.y, dstOut + pos); ++pos; }
        if (p1) { __builtin_nontemporal_store((float)eA.z, srcOut + pos);
                  __builtin_nontemporal_store((float)eA.w, dstOut + pos); ++pos; }
        if (p2) { __builtin_nontemporal_store((float)eB.x, srcOut + pos);
                  __builtin_nontemporal_store((float)eB.y, dstOut + pos); ++pos; }
        if (p3) { __builtin_nontemporal_store((float)eB.z, srcOut + pos);
                  __builtin_nontemporal_store((float)eB.w, dstOut + pos); ++pos; }
    }
}

// ---------------------------------------------------------------------------
// Launch: inputs are {distanceVector: float[E], edgeCands: int[E,2]}.
// d_out = [src(E) | dst(E) | n_edges(1)] as float. d_ws: NBLK u32 counts.
// ---------------------------------------------------------------------------
extern "C" void kernel_launch(void* const* d_in, const int* in_sizes, int n_in,
                              void* d_out, int out_size, void* d_ws, size_t ws_size,
                              hipStream_t stream) {
    const float* dist  = (const float*)d_in[0];
    const int*   edges = (const int*)d_in[1];
    float*       out   = (float*)d_out;
    unsigned*    blk   = (unsigned*)d_ws;   // NBLK * 4B = 128 KB scratch

    k_count_fill<<<NBLK, BLOCK, 0, stream>>>(dist, out, blk);
    k_scan<<<1, SCAN_T, 0, stream>>>(blk, out);
    k_scatter<<<NBLK, BLOCK, 0, stream>>>(dist, edges, blk, out);
}